// BatchedActivationCSA_29188597743843
// MI455X (gfx1250) — compile-verified
//
#include <hip/hip_runtime.h>
#include <hip/hip_bf16.h>

// ---------------------------------------------------------------------------
// BatchedActivationCSA for MI455X (gfx1250, wave32)
//
//   BSZ=4, SEQ=4096, DIM=1024, MEASURE_DIM=64, TOPK=16, NUM_TARGETS=1
//
// Math (per row): fold 1024->64 bins via perm/sign gather; z = H64*s/32;
// gate, exact top-16 + tau threshold, *alpha; w = H64*zs/32;
// out[d] = x[d] + signs[d]*w[inv_perm[d]&63].
//
// The two 64-pt Hadamards run as f32 WMMA GEMMs (V_WMMA_F32_16X16X4_F32)
// against the +/-1 Hadamard matrix synthesized from popcount(row&col).
// ---------------------------------------------------------------------------

#define DIM        1024
#define MD         64
#define ROWS_TOTAL (4 * 4096)
#define SBS        65                 // padded LDS row stride (bank-conflict free)
#define WPB        4                  // waves per block
#define RPW        16                 // rows per wave (one WMMA M tile)
#define RPB        (WPB * RPW)        // 64 rows per block

typedef __attribute__((ext_vector_type(2))) float v2f;
typedef __attribute__((ext_vector_type(8))) float v8f;

// D(16x16) += A(16x4) x B(4x16), f32. Layouts per CDNA5 ISA 7.12.2:
//  A: lane l<16 -> row l, (K0,K1); lane l+16 -> row l, (K2,K3)
//  B: lanes 0-15 -> K rows {r0,r0+1}, col = lane; lanes 16-31 -> same (r0=+2 half)
//  D: VGPR v: lanes 0-15 -> M=v, N=lane; lanes 16-31 -> M=v+8, N=lane-16
__device__ __forceinline__ v8f wmma_f32_4(v2f a, v2f b, v8f c) {
    return __builtin_amdgcn_wmma_f32_16x16x4_f32(false, a, false, b,
                                                 (short)0, c, false, false);
}

// In-place: S(16 rows x 64 cols, stride SBS in LDS) <- (gate?) * (S x H64) / 32
template <bool GATED>
__device__ __forceinline__ void fwht64_wmma(float* __restrict__ S,
                                            const float* __restrict__ gates,
                                            int b, int lane) {
    const int half = lane >> 4;
    const int lr   = lane & 15;

    v8f acc[4];
    #pragma unroll
    for (int n = 0; n < 4; ++n) acc[n] = (v8f){0.f,0.f,0.f,0.f,0.f,0.f,0.f,0.f};

    #pragma unroll 4
    for (int kk = 0; kk < 16; ++kk) {
        const int r0 = 4 * kk + 2 * half;      // this lane-half's K rows: r0, r0+1
        v2f a;
        a.x = S[lr * SBS + r0];
        a.y = S[lr * SBS + r0 + 1];
        #pragma unroll
        for (int n = 0; n < 4; ++n) {
            const int col = 16 * n + lr;
            v2f bb;                            // Hadamard entries, +/-1 (symmetric)
            bb.x = (__builtin_popcount((unsigned)(r0 & col)) & 1) ? -1.0f : 1.0f;
            bb.y = (__builtin_popcount((unsigned)((r0 + 1) & col)) & 1) ? -1.0f : 1.0f;
            acc[n] = wmma_f32_4(a, bb, acc[n]);
        }
    }

    #pragma unroll
    for (int n = 0; n < 4; ++n) {
        const int col = 16 * n + lr;
        float scale = 0.03125f;                // 1/sqrt(1024)
        if (GATED) scale *= gates[b * MD + col];
        #pragma unroll
        for (int v = 0; v < 8; ++v) {
            S[(v + 8 * half) * SBS + col] = acc[n][v] * scale;
        }
    }
}

__global__ __launch_bounds__(WPB * 32) void csa_kernel(
    const float* __restrict__ x, const float* __restrict__ gates,
    const float* __restrict__ alpha, const float* __restrict__ tau,
    const float* __restrict__ signs, const int* __restrict__ perm,
    const int* __restrict__ inv_perm, float* __restrict__ out) {

    __shared__ int   ls_gidx[DIM];     // perm[p]
    __shared__ float ls_gsgn[DIM];     // signs[perm[p]]
    __shared__ int   ls_cidx[DIM];     // inv_perm[d] & 63
    __shared__ float ls_sign[DIM];     // signs[d]
    __shared__ float ls_sb[WPB][RPW * SBS];

    const int tid = threadIdx.x;
    for (int i = tid; i < DIM; i += WPB * 32) {
        const int p = perm[i];
        ls_gidx[i] = p;
        ls_gsgn[i] = signs[p];
        ls_cidx[i] = inv_perm[i] & (MD - 1);
        ls_sign[i] = signs[i];
    }
    __syncthreads();

    const int lane    = tid & 31;
    const int wv      = tid >> 5;
    const int rowbase = blockIdx.x * RPB + wv * RPW;
    const int b       = rowbase >> 12;           // SEQ = 4096 rows per batch
    float* sb = ls_sb[wv];

    // ---- Stage 1: fold 1024 -> 64 bins per row (signed permuted gather) ----
    for (int i = 0; i < 32; ++i) {
        const int idx = i * 32 + lane;           // (row, bin) pair, 16*64 total
        const int r = idx >> 6, m = idx & 63;
        const float* xr = x + (size_t)(rowbase + r) * DIM;
        float s = 0.f;
        #pragma unroll
        for (int h = 0; h < 16; ++h) {
            const int p = h * MD + m;
            s += xr[ls_gidx[p]] * ls_gsgn[p];
        }
        sb[r * SBS + m] = s;
    }
    __syncthreads();

    // ---- Stage 2: z = gate * (S x H64)/32, via f32 WMMA ----
    fwht64_wmma<true>(sb, gates, b, lane);
    __syncthreads();

    // ---- Stage 3: exact top-16 + tau threshold + alpha (1 lane per row) ----
    if (lane < RPW) {
        const float tauv = __builtin_fabsf(tau[b]);
        const float av   = alpha[b];
        float* R = sb + lane * SBS;
        unsigned long long selmask = 0ull;
        #pragma unroll 1
        for (int t = 0; t < 16; ++t) {
            float bestv = -1.f;
            int   bestm = 0;
            #pragma unroll 1
            for (int m = 0; m < MD; ++m) {
                const bool sel = (selmask >> m) & 1ull;
                const float v = __builtin_fabsf(R[m]);
                if (!sel && v > bestv) { bestv = v; bestm = m; }  // low-idx tiebreak
            }
            selmask |= (1ull << bestm);
        }
        #pragma unroll 1
        for (int m = 0; m < MD; ++m) {
            const float g = R[m];
            const bool keep = ((selmask >> m) & 1ull) && (__builtin_fabsf(g) >= tauv);
            R[m] = keep ? av * g : 0.f;
        }
    }
    __syncthreads();

    // ---- Stage 4: w = (ZS x H64)/32, via f32 WMMA ----
    fwht64_wmma<false>(sb, gates, b, lane);
    __syncthreads();

    // ---- Stage 5: out[d] = x[d] + signs[d] * w[inv_perm[d]&63], float4 ----
    for (int i = 0; i < (RPW * DIM) / (32 * 4); ++i) {
        const int e = (i * 32 + lane) * 4;
        const int r = e >> 10, d = e & (DIM - 1);
        const size_t off = (size_t)(rowbase + r) * DIM + d;
        const float4 xv = *(const float4*)(x + off);
        const int4   cv = *(const int4*)(ls_cidx + d);
        const float4 sv = *(const float4*)(ls_sign + d);
        const float* W  = sb + r * SBS;
        float4 ov;
        ov.x = xv.x + sv.x * W[cv.x];
        ov.y = xv.y + sv.y * W[cv.y];
        ov.z = xv.z + sv.z * W[cv.z];
        ov.w = xv.w + sv.w * W[cv.w];
        *(float4*)(out + off) = ov;
    }
}

extern "C" void kernel_launch(void* const* d_in, const int* in_sizes, int n_in,
                              void* d_out, int out_size, void* d_ws, size_t ws_size,
                              hipStream_t stream) {
    const float* x        = (const float*)d_in[0];
    const float* gates    = (const float*)d_in[1];
    const float* alpha    = (const float*)d_in[2];
    const float* tau      = (const float*)d_in[3];
    const float* signs    = (const float*)d_in[4];
    const int*   perm     = (const int*)d_in[5];
    const int*   inv_perm = (const int*)d_in[6];
    // d_in[7] = target_idx (always 0: NUM_TARGETS == 1)

    const int blocks = ROWS_TOTAL / RPB;   // 16384 / 64 = 256
    csa_kernel<<<dim3(blocks), dim3(WPB * 32), 0, stream>>>(
        x, gates, alpha, tau, signs, perm, inv_perm, (float*)d_out);
}